// GroupRelationAggregator_30648886624754
// MI455X (gfx1250) — compile-verified
//
#include <hip/hip_runtime.h>

typedef float v2f __attribute__((ext_vector_type(2)));
typedef float v8f __attribute__((ext_vector_type(8)));

#define BATCH 2
#define NPTS  65536
#define KNBR  16
#define DIMF  35
#define OUTC  128
#define GPB   (NPTS / 16)   // point-groups per batch

__device__ __forceinline__ v8f wmma4(v2f a, v2f b, v8f c) {
  // D = A(16x4,f32) * B(4x16,f32) + C(16x16,f32)
  return __builtin_amdgcn_wmma_f32_16x16x4_f32(false, a, false, b, (short)0, c,
                                               false, false);
}

__device__ __forceinline__ float bn_scale(const float* p, int c, int i) {
  return p[i] * rsqrtf(p[3 * c + i] + 1e-5f);
}
__device__ __forceinline__ float bn_shift(const float* p, int c, int i, float sc) {
  return p[c + i] - p[2 * c + i] * sc;
}

__global__ __launch_bounds__(256)
void gra_kernel(const float* __restrict__ feats, const int* __restrict__ idx,
                const float* __restrict__ pW, const float* __restrict__ pBN,
                const float* __restrict__ eW, const float* __restrict__ eBN,
                const float* __restrict__ mW, const float* __restrict__ mBN,
                const float* __restrict__ gW, const float* __restrict__ gBN,
                const float* __restrict__ mapW, const float* __restrict__ mapBN,
                const float* __restrict__ fcW, float* __restrict__ out)
{
  // block-shared folded weights
  __shared__ __align__(16) float sB[48 * 48];    // combined B, col-major: sB[c*48+k]
  __shared__ __align__(16) float sFC[128 * 32];  // fc_W transposed: sFC[n*32+k]
  __shared__ __align__(16) float sWE[32];        // eta fold vector
  __shared__ __align__(16) float sW2[16];        // per-column map weights for tile2
  __shared__ float sC0;
  // per-wave scratch (8 waves)
  __shared__ __align__(16) float sA[8][16 * 48]; // A matrix, 16 rows x 48 cols
  __shared__ __align__(16) float sV[8][16 * 32]; // pre-fc vectors, 16 pts x 32 ch

  const int tid  = threadIdx.x;
  const int lane = tid & 31;
  const int w    = tid >> 5;

  const float map_sc = bn_scale(mapBN, 1, 0);
  const float map_sh = bn_shift(mapBN, 1, 0, map_sc);

  // ---- build combined 48x48 B (gamma | mu | point), BN folded, bias in row 45 ----
  for (int t = tid; t < 48 * 48; t += 256) {
    const int c = t / 48, k = t % 48;
    float val = 0.f;
    if (c < 32) {                       // gamma: A cols 0..34 (full nbr row)
      const float sc = bn_scale(gBN, 32, c);
      if (k < 35)       val = gW[k * 32 + c] * sc;
      else if (k == 45) val = bn_shift(gBN, 32, c, sc);
    } else if (c < 40) {                // mu: A cols 3..34 (pnf)
      const int cc = c - 32;
      const float sc = bn_scale(mBN, 8, cc);
      if (k >= 3 && k < 35)  val = mW[(k - 3) * 8 + cc] * sc;
      else if (k == 45)      val = bn_shift(mBN, 8, cc, sc);
    } else {                            // point: A cols 35..44 (point10)
      const int cc = c - 40;
      const float sc = bn_scale(pBN, 8, cc);
      if (k >= 35 && k < 45) val = pW[(k - 35) * 8 + cc] * sc;
      else if (k == 45)      val = bn_shift(pBN, 8, cc, sc);
    }
    sB[c * 48 + k] = val;
  }
  // ---- fc_W transposed ----
  for (int t = tid; t < 128 * 32; t += 256) {
    const int n = t >> 5, k = t & 31;
    sFC[t] = fcW[k * 128 + n];
  }
  // ---- eta fold: we_fold[i] = sum_c eta_W[i][c]*esc[c]*wm[8+c] ----
  if (tid < 32) {
    float s = 0.f;
    for (int c = 0; c < 8; ++c) {
      const float esc = bn_scale(eBN, 8, c);
      s += eW[tid * 8 + c] * esc * (mapW[8 + c] * map_sc);
    }
    sWE[tid] = s;
  }
  // ---- tile2 per-column map weights: cols 0..7=mu -> wm[16+c], 8..15=point -> wm[c-8]
  if (tid < 16)
    sW2[tid] = (tid < 8) ? (mapW[16 + tid] * map_sc) : (mapW[tid - 8] * map_sc);
  if (tid == 0) {
    float s = map_sh;
    for (int c = 0; c < 8; ++c) {
      const float esc = bn_scale(eBN, 8, c);
      s += bn_shift(eBN, 8, c, esc) * (mapW[8 + c] * map_sc);
    }
    sC0 = s;
  }
  __syncthreads();

  const int   m  = lane & 15;     // A/D row-half index & B/D column index
  const int   h  = lane >> 4;     // K-half select
  const float weL = sWE[lane];
  const float w2L = sW2[m];
  const bool  pointHalf = (m >= 8);
  const float C0 = sC0;

  // ---- persistent combined-B WMMA operands (3 n-tiles x 12 k-steps = 72 VGPRs) ----
  v2f bg[3][12];
  #pragma unroll
  for (int nt = 0; nt < 3; ++nt)
    #pragma unroll
    for (int kk = 0; kk < 12; ++kk)
      bg[nt][kk] = *(const v2f*)&sB[(nt * 16 + m) * 48 + 4 * kk + 2 * h];

  const int waveId  = blockIdx.x * 8 + w;
  const int nWaves  = gridDim.x * 8;
  const int nGroups = BATCH * GPB;

  for (int g = waveId; g < nGroups; g += nWaves) {
    const int b  = g / GPB;
    const int n0 = (g % GPB) * 16;
    float* Aw = &sA[w][0];

    // ===================== phase 1: one point per iteration =====================
    for (int p = 0; p < 16; ++p) {
      const int  n      = n0 + p;
      const long rowOwn = (long)b * NPTS + n;
      const int  j      = idx[rowOwn * KNBR + m];
      const long base   = ((long)b * NPTS + j) * DIMF;

      // gather neighbor row m: half 0 -> cols 0..17, half 1 -> cols 18..34
      float pnx = 0.f, pny = 0.f, pnz = 0.f;
      #pragma unroll
      for (int i = 0; i < 18; ++i) {
        const int col = h * 18 + i;
        if (col < DIMF) {
          const float x = feats[base + col];
          Aw[m * 48 + col] = x;
          if (h == 0) {
            if (i == 0) pnx = x; else if (i == 1) pny = x; else if (i == 2) pnz = x;
          }
        }
      }
      const float* fo = feats + rowOwn * DIMF;
      const float px = fo[0], py = fo[1], pz = fo[2];
      const float featOwn = fo[3 + lane];
      if (h == 0) {  // build point10 | 1 | 0,0 (cols 35..47)
        const float dx = px - pnx, dy = py - pny, dz = pz - pnz;
        float* row = &Aw[m * 48];
        row[35] = dx * dx + dy * dy + dz * dz;
        row[36] = px;  row[37] = py;  row[38] = pz;
        row[39] = pnx; row[40] = pny; row[41] = pnz;
        row[42] = dx;  row[43] = dy;  row[44] = dz;
        row[45] = 1.f; row[46] = 0.f; row[47] = 0.f;
      }
      asm volatile("s_wait_dscnt 0" ::: "memory");  // cross-lane LDS hand-off

      // A operands: lane m = neighbor row, K pair (2h, 2h+1) per k-step
      v2f a[12];
      #pragma unroll
      for (int kk = 0; kk < 12; ++kk)
        a[kk] = *(const v2f*)&Aw[m * 48 + 4 * kk + 2 * h];

      v8f d0 = {0,0,0,0,0,0,0,0}, d1 = d0, d2t = d0;
      #pragma unroll
      for (int kk = 0; kk < 12; ++kk) {
        d0  = wmma4(a[kk], bg[0][kk], d0);   // gam cols  0..15
        d1  = wmma4(a[kk], bg[1][kk], d1);   // gam cols 16..31
        d2t = wmma4(a[kk], bg[2][kk], d2t);  // mu(0..7) | point_pre(8..15)
      }

      // etadot = feat . we_fold (full-wave reduction)
      float e = featOwn * weL;
      e += __shfl_xor(e, 1, 32);  e += __shfl_xor(e, 2, 32);
      e += __shfl_xor(e, 4, 32);  e += __shfl_xor(e, 8, 32);
      e += __shfl_xor(e, 16, 32);

      // per-neighbor scalar s and max-over-K epilogue on D-tile layout
      float vm0 = -3.4e38f, vm1 = -3.4e38f;
      #pragma unroll
      for (int r = 0; r < 8; ++r) {
        float t = d2t[r];
        if (pointHalf) t = fmaxf(t, 0.f);   // relu on point branch only
        t *= w2L;
        t += __shfl_xor(t, 1, 32); t += __shfl_xor(t, 2, 32);
        t += __shfl_xor(t, 4, 32); t += __shfl_xor(t, 8, 32);
        const float s = fmaxf(t + e + C0, 0.f);   // map relu
        vm0 = fmaxf(vm0, s * d0[r]);
        vm1 = fmaxf(vm1, s * d1[r]);
      }
      vm0 = fmaxf(vm0, __shfl_xor(vm0, 16, 32));  // fold neighbor halves
      vm1 = fmaxf(vm1, __shfl_xor(vm1, 16, 32));
      if (lane < 16) {
        sV[w][p * 32 + lane]      = vm0;
        sV[w][p * 32 + 16 + lane] = vm1;
      }
    }
    asm volatile("s_wait_dscnt 0" ::: "memory");

    // ===================== phase 2: batched 16x32x128 FC =====================
    v2f fa[8];
    #pragma unroll
    for (int kk = 0; kk < 8; ++kk)
      fa[kk] = *(const v2f*)&sV[w][m * 32 + 4 * kk + 2 * h];

    #pragma unroll
    for (int nt = 0; nt < 8; ++nt) {
      v8f acc = {0,0,0,0,0,0,0,0};
      #pragma unroll
      for (int kk = 0; kk < 8; ++kk) {
        const v2f fb = *(const v2f*)&sFC[(nt * 16 + m) * 32 + 4 * kk + 2 * h];
        acc = wmma4(fa[kk], fb, acc);
      }
      #pragma unroll
      for (int r = 0; r < 8; ++r)
        out[((long)b * NPTS + n0 + r + 8 * h) * OUTC + nt * 16 + m] = acc[r];
    }
  }
}

extern "C" void kernel_launch(void* const* d_in, const int* in_sizes, int n_in,
                              void* d_out, int out_size, void* d_ws, size_t ws_size,
                              hipStream_t stream) {
  const float* feats = (const float*)d_in[0];
  const int*   idx   = (const int*)  d_in[1];
  const float* pW    = (const float*)d_in[2];
  const float* pBN   = (const float*)d_in[3];
  const float* eW    = (const float*)d_in[4];
  const float* eBN   = (const float*)d_in[5];
  const float* mW    = (const float*)d_in[6];
  const float* mBN   = (const float*)d_in[7];
  const float* gW    = (const float*)d_in[8];
  const float* gBN   = (const float*)d_in[9];
  const float* mapW  = (const float*)d_in[10];
  const float* mapBN = (const float*)d_in[11];
  const float* fcW   = (const float*)d_in[12];
  float* out = (float*)d_out;

  gra_kernel<<<512, 256, 0, stream>>>(feats, idx, pW, pBN, eW, eBN, mW, mBN,
                                      gW, gBN, mapW, mapBN, fcW, out);
}